// Convert_48284022342006
// MI455X (gfx1250) — compile-verified
//
#include <hip/hip_runtime.h>

// CDNA5 / gfx1250: fp32 WMMA 16x16x4 for the per-batch 40x40x40 GEMMs.
typedef __attribute__((ext_vector_type(2))) float v2f;
typedef __attribute__((ext_vector_type(8))) float v8f;

#define P   40
#define PT  48           // padded to 3x16 tiles
#define MST 50           // md2 row stride in dwords: gcd(50,64)=2 -> conflict-free, 8B aligned
#define WST 42           // transposed-w row stride: gcd(42,64)=2 -> conflict-free, 8B aligned
#define NF  7
#define WPB 4            // waves per block (wave32)
#define BLOCK (WPB * 32)
#define FLTMAX 3.402823466e+38f

#define WMMA_F32X4(A, Bv, C) \
  __builtin_amdgcn_wmma_f32_16x16x4_f32(false, (A), false, (Bv), (short)0, (C), false, false)

__global__ __launch_bounds__(BLOCK)
void lorentz_feats_kernel(const float* __restrict__ x,   // [B][4][P]
                          const float* __restrict__ w,   // [5][P][P]
                          float* __restrict__ out,       // [B][NF][P]
                          int B)
{
  // block-shared weights
  __shared__ __align__(16) float w0s[P][P];       // w[0] for Es @ w0
  __shared__ float ws1[P];                        // rowsum(w[1])
  __shared__ float ws2[P];                        // rowsum(w[2])
  __shared__ __align__(16) float w3t[PT][WST];    // w3t[n][i] = w[3][i][n]
  __shared__ __align__(16) float w4t[PT][WST];    // w4t[n][i] = w[4][i][n]
  // per-wave scratch
  __shared__ __align__(16) float md2[WPB][PT][MST]; // symmetric Minkowski matrix
  __shared__ float xsh[WPB][4][P];
  __shared__ float ssh[WPB][P];                   // s_i = -Ms_i

  const int tid  = threadIdx.x;
  const int lane = tid & 31;
  const int wv   = tid >> 5;

  // ---- stage weights in LDS (once) ----
  for (int idx = tid; idx < P * P; idx += BLOCK) {
    int i = idx / P, n = idx % P;
    w0s[i][n] = w[0 * P * P + idx];
    w3t[n][i] = w[3 * P * P + idx];   // transpose: K-adjacent per lane
    w4t[n][i] = w[4 * P * P + idx];
  }
  for (int idx = tid; idx < (PT - P) * WST; idx += BLOCK) {
    int n = P + idx / WST, i = idx % WST;
    w3t[n][i] = 0.f;
    w4t[n][i] = 0.f;
  }
  for (int i = tid; i < P; i += BLOCK) {
    float s1 = 0.f, s2 = 0.f;
    for (int k = 0; k < P; ++k) {
      s1 += w[1 * P * P + i * P + k];
      s2 += w[2 * P * P + i * P + k];
    }
    ws1[i] = s1;
    ws2[i] = s2;
  }
  __syncthreads();

  // WMMA f32 16x16x4 operand indexing (wave32):
  //  A (16x4): lanes 0-15 -> K=0/1 in .x/.y ; lanes 16-31 -> K=2/3
  //  B (4x16): same K split; N = lane&15
  //  D: VGPR r holds row (r + 8*half), col = lane&15
  const int half = lane >> 4;
  const int ln   = lane & 15;
  const int kb   = half << 1;

  // ---- cache ALL B fragments in VGPRs: invariant across the whole batch ----
  v2f bw3[3][10], bw4[3][10];   // [n-tile][k-step], 120 VGPRs
  #pragma unroll
  for (int t = 0; t < 3; ++t) {
    #pragma unroll
    for (int ks = 0; ks < 10; ++ks) {
      bw3[t][ks] = *(const v2f*)&w3t[16 * t + ln][4 * ks + kb];  // ds_load_b64
      bw4[t][ks] = *(const v2f*)&w4t[16 * t + ln][4 * ks + kb];
    }
  }

  const int gwave  = blockIdx.x * WPB + wv;
  const int nwaves = gridDim.x * WPB;

  for (int b = gwave; b < B; b += nwaves) {
    const float* xb = x + (size_t)b * 4 * P;
    if (b + nwaves < B)
      __builtin_prefetch(x + (size_t)(b + nwaves) * 4 * P, 0, 3); // global_prefetch_b8

    // ---- load x[b], compute s_i ----
    for (int i = lane; i < P; i += 32) {
      float x0 = xb[0 * P + i], x1 = xb[1 * P + i];
      float x2 = xb[2 * P + i], x3 = xb[3 * P + i];
      xsh[wv][0][i] = x0; xsh[wv][1][i] = x1;
      xsh[wv][2][i] = x2; xsh[wv][3][i] = x3;
      ssh[wv][i] = -x0 * x0 + x1 * x1 + x2 * x2 + x3 * x3;  // = -Ms
    }
    __builtin_amdgcn_wave_barrier();

    // ---- build symmetric md2 (40x40 live region; 50 uniform iters/lane) ----
    for (int idx = lane; idx < P * P; idx += 32) {
      int i = idx / P, j = idx % P;
      float dot = -xsh[wv][0][i] * xsh[wv][0][j]
                 + xsh[wv][1][i] * xsh[wv][1][j]
                 + xsh[wv][2][i] * xsh[wv][2][j]
                 + xsh[wv][3][i] * xsh[wv][3][j];
      md2[wv][i][j] = ssh[wv][i] + ssh[wv][j] - 2.f * dot;
    }
    __builtin_amdgcn_wave_barrier();

    float* ob = out + (size_t)b * NF * P;

    // ---- min features: T = md2 @ w3 / w4 via v_wmma_f32_16x16x4_f32 ----
    // Rows j>=40 of A read stale LDS; their D rows are discarded (row-confined).
    for (int j0 = 0; j0 < PT; j0 += 16) {
      v8f c3a = {}, c3b = {}, c3c = {};
      v8f c4a = {}, c4b = {}, c4c = {};
      #pragma unroll
      for (int ks = 0; ks < 10; ++ks) {
        v2f a = *(const v2f*)&md2[wv][j0 + ln][4 * ks + kb];   // ds_load_b64
        c3a = WMMA_F32X4(a, bw3[0][ks], c3a);
        c3b = WMMA_F32X4(a, bw3[1][ks], c3b);
        c3c = WMMA_F32X4(a, bw3[2][ks], c3c);
        c4a = WMMA_F32X4(a, bw4[0][ks], c4a);
        c4b = WMMA_F32X4(a, bw4[1][ks], c4b);
        c4c = WMMA_F32X4(a, bw4[2][ks], c4c);
      }

      // min over k (cols): tile 2 only has 8 valid cols (k = 32..39)
      #pragma unroll
      for (int r = 0; r < 8; ++r) {
        int j = j0 + (half << 3) + r;
        float m3 = fminf(c3a[r], c3b[r]);
        m3 = fminf(m3, (ln < 8) ? c3c[r] : FLTMAX);
        float m4 = fminf(c4a[r], c4b[r]);
        m4 = fminf(m4, (ln < 8) ? c4c[r] : FLTMAX);
        #pragma unroll
        for (int off = 1; off <= 8; off <<= 1) {   // reduce within 16-lane half
          m3 = fminf(m3, __shfl_xor(m3, off, 32));
          m4 = fminf(m4, __shfl_xor(m4, off, 32));
        }
        if (ln == 0 && j < P) {
          ob[5 * P + j] = m3;
          ob[6 * P + j] = m4;
        }
      }
    }

    // ---- simple + sum features (VALU matvecs) ----
    for (int j = lane; j < P; j += 32) {
      float x0 = xsh[wv][0][j], x1 = xsh[wv][1][j];
      float x2 = xsh[wv][2][j], x3 = xsh[wv][3][j];
      ob[0 * P + j] = x0 * x0 - x1 * x1 - x2 * x2 - x3 * x3;   // Ms
      ob[1 * P + j] = sqrtf(x1 * x1 + x2 * x2);                // Pts
      float a0 = 0.f, a1 = 0.f, a2 = 0.f;
      for (int i = 0; i < P; ++i) {
        a0 += xsh[wv][0][i] * w0s[i][j];       // Es @ w0
        float m = md2[wv][i][j];
        a1 += m * ws1[i];                      // (md2 @ w1).sum(axis=k)
        a2 += m * ws2[i];                      // (md2 @ w2).sum(axis=k)
      }
      ob[2 * P + j] = a0;
      ob[3 * P + j] = a1;
      ob[4 * P + j] = a2;
    }
    __builtin_amdgcn_wave_barrier();
  }
}

extern "C" void kernel_launch(void* const* d_in, const int* in_sizes, int n_in,
                              void* d_out, int out_size, void* d_ws, size_t ws_size,
                              hipStream_t stream) {
  const float* x = (const float*)d_in[0];
  const float* w = (const float*)d_in[1];
  float* out = (float*)d_out;
  const int B = in_sizes[0] / (4 * P);   // 16384 for the reference shapes

  int totalWaves = (B + 3) / 4;          // ~4 batch elements per wave
  int grid = (totalWaves + WPB - 1) / WPB;
  if (grid > 1024) grid = 1024;
  if (grid < 1) grid = 1;

  lorentz_feats_kernel<<<grid, BLOCK, 0, stream>>>(x, w, out, B);
}